// NGCF_25589415150204
// MI455X (gfx1250) — compile-verified
//
#include <hip/hip_runtime.h>

#define NUM_USERS 50000
#define NUM_ITEMS 100000
#define N_NODES   150000
#define NNZ       2400000
#define EMBED     64
#define N_LAYERS  3
#define BATCH     4096
#define NEG_SLOPE 0.2f
#define EPS_NORM  1e-12f

typedef __attribute__((ext_vector_type(2))) float v2f;
typedef __attribute__((ext_vector_type(8))) float v8f;

__device__ __forceinline__ void atomic_add_f32(float* p, float v) {
  // relaxed, device scope -> global_atomic_add_f32 (no CAS loop)
  __hip_atomic_fetch_add(p, v, __ATOMIC_RELAXED, __HIP_MEMORY_SCOPE_AGENT);
}

// ---------------------------------------------------------------------------
// E0 = concat(embed_user, embed_item), float4-vectorized
// ---------------------------------------------------------------------------
__global__ void k_init_E(const float4* __restrict__ eu,
                         const float4* __restrict__ ei,
                         float4* __restrict__ E) {
  int t = blockIdx.x * blockDim.x + threadIdx.x;   // N_NODES*16 float4s
  const int nu4 = NUM_USERS * 16;
  const int tot = N_NODES * 16;
  if (t >= tot) return;
  E[t] = (t < nu4) ? eu[t] : ei[t - nu4];
}

__global__ void k_zero(float4* __restrict__ p, int n4) {
  int t = blockIdx.x * blockDim.x + threadIdx.x;
  if (t < n4) p[t] = make_float4(0.f, 0.f, 0.f, 0.f);
}

// ---------------------------------------------------------------------------
// LE[row] += val * E[col]  — one wave per edge, float2 per lane, L2 atomics.
// Edge metadata is wave-uniform -> readfirstlane -> scalar (s_load) path,
// keeping VMEM free for the E gather + atomic stream that bound this phase.
// ---------------------------------------------------------------------------
__global__ void __launch_bounds__(256) k_scatter(
    const float* __restrict__ eval, const int* __restrict__ erow,
    const int* __restrict__ ecol, const float* __restrict__ E,
    float* __restrict__ LE) {
  int t = blockIdx.x * blockDim.x + threadIdx.x;   // NNZ*32 threads (exact)
  int e = __builtin_amdgcn_readfirstlane(t >> 5);  // wave-uniform edge id
  int c = (threadIdx.x & 31) * 2;
  if (e >= NNZ) return;
  float v  = eval[e];   // scalar loads (uniform index)
  int col  = ecol[e];
  int row  = erow[e];
  const float2 s = *(const float2*)(E + (long long)col * EMBED + c);
  float* dst = LE + (long long)row * EMBED + c;
  atomic_add_f32(dst + 0, v * s.x);
  atomic_add_f32(dst + 1, v * s.y);
}

// ---------------------------------------------------------------------------
// E_next = L2normalize(leaky_relu((LE+E)@W1^T + b1 + LE@W2^T + b2))
// One wave per 16-row tile. fp32 WMMA 16x16x4, K=64 in 16 steps.
// Col-tile outermost; all 32 B fragments of a col-tile preloaded into
// registers so the 32 WMMAs issue back-to-back against LDS A fragments.
// ---------------------------------------------------------------------------
__global__ void __launch_bounds__(32) k_layer(
    const float* __restrict__ LE, const float* __restrict__ E,
    const float* __restrict__ W1, const float* __restrict__ b1,
    const float* __restrict__ W2, const float* __restrict__ b2,
    float* __restrict__ Eout) {
  __shared__ float sX1[16 * 68];   // LE+E tile (stride 68 -> conflict-free)
  __shared__ float sX2[16 * 68];   // LE tile
  __shared__ float sOut[16 * 68];  // activated D tile
  __shared__ float sScale[16];

  const int lane = threadIdx.x;                           // 0..31, full wave
  const long long rowbase = (long long)blockIdx.x * 16;   // exact: 150000/16

  // Stage X tiles: 16x64 floats each, 8 float4s per lane
  #pragma unroll
  for (int j = 0; j < 8; ++j) {
    int flat = j * 128 + lane * 4;          // 0..1023
    int m = flat >> 6, c = flat & 63;
    const float4 le4 = *(const float4*)(LE + (rowbase + m) * 64 + c);
    const float4 e4  = *(const float4*)(E  + (rowbase + m) * 64 + c);
    *(float4*)(sX2 + m * 68 + c) = le4;
    float4 x1 = make_float4(le4.x + e4.x, le4.y + e4.y, le4.z + e4.z, le4.w + e4.w);
    *(float4*)(sX1 + m * 68 + c) = x1;
  }
  __syncthreads();

  const int mloc = lane & 15;
  const int koff = (lane >> 4) * 2;         // lanes<16: K=0,1 ; lanes>=16: K=2,3
  const int nloc = lane & 15;
  const int hi   = lane >> 4;

  #pragma unroll
  for (int t = 0; t < 4; ++t) {
    const int n = t * 16 + nloc;

    // Preload all B fragments of this col-tile (batched loads, one wait)
    v2f bw1[16], bw2[16];
    #pragma unroll
    for (int kk = 0; kk < 16; ++kk) {
      bw1[kk] = *(const v2f*)(W1 + n * 64 + kk * 4 + koff);  // B[K][N]=W[n][k]
      bw2[kk] = *(const v2f*)(W2 + n * 64 + kk * 4 + koff);
    }

    v8f acc = (v8f){0.f, 0.f, 0.f, 0.f, 0.f, 0.f, 0.f, 0.f};
    #pragma unroll
    for (int kk = 0; kk < 16; ++kk) {
      const int kb = kk * 4 + koff;
      v2f a1 = *(const v2f*)(sX1 + mloc * 68 + kb);
      v2f a2 = *(const v2f*)(sX2 + mloc * 68 + kb);
      acc = __builtin_amdgcn_wmma_f32_16x16x4_f32(
          false, a1, false, bw1[kk], (short)0, acc, false, false);
      acc = __builtin_amdgcn_wmma_f32_16x16x4_f32(
          false, a2, false, bw2[kk], (short)0, acc, false, false);
    }

    // bias + leaky relu, stage D col-tile into sOut in plain [16][64] layout
    const float bias = b1[n] + b2[n];
    #pragma unroll
    for (int r = 0; r < 8; ++r) {
      float x = acc[r] + bias;
      x = (x > 0.f) ? x : NEG_SLOPE * x;
      sOut[(r + hi * 8) * 68 + n] = x;
    }
  }
  __syncthreads();

  // row L2 norms (16 rows, lanes 0..15)
  if (lane < 16) {
    float s = 0.f;
    #pragma unroll
    for (int c = 0; c < 64; ++c) {
      float x = sOut[lane * 68 + c];
      s += x * x;
    }
    sScale[lane] = 1.0f / fmaxf(sqrtf(s), EPS_NORM);
  }
  __syncthreads();

  // normalized store
  #pragma unroll
  for (int j = 0; j < 8; ++j) {
    int flat = j * 128 + lane * 4;
    int m = flat >> 6, c = flat & 63;
    const float sc = sScale[m];
    float4 x = *(const float4*)(sOut + m * 68 + c);
    float4 o = make_float4(x.x * sc, x.y * sc, x.z * sc, x.w * sc);
    *(float4*)(Eout + (rowbase + m) * 64 + c) = o;
  }
}

// ---------------------------------------------------------------------------
// Gather layer-l embeddings of the three batches into out[:, l*64:(l+1)*64]
// out = [user(4096x256) | pos(4096x256) | neg(4096x256)] flat
// ---------------------------------------------------------------------------
__global__ void k_gather(const float* __restrict__ El, int l,
                         const int* __restrict__ bu, const int* __restrict__ bp,
                         const int* __restrict__ bn, float* __restrict__ out) {
  int t = blockIdx.x * blockDim.x + threadIdx.x;   // 3*4096*16
  if (t >= 3 * BATCH * 16) return;
  int c4  = t & 15;
  int b   = (t >> 4) & (BATCH - 1);
  int seg = t >> 16;                               // BATCH*16 == 65536
  int node;
  if (seg == 0)      node = bu[b];
  else if (seg == 1) node = NUM_USERS + bp[b];
  else               node = NUM_USERS + bn[b];
  const float4 v = *(const float4*)(El + (long long)node * 64 + c4 * 4);
  *(float4*)(out + (long long)seg * BATCH * 256 + (long long)b * 256 + l * 64 + c4 * 4) = v;
}

// ---------------------------------------------------------------------------
extern "C" void kernel_launch(void* const* d_in, const int* in_sizes, int n_in,
                              void* d_out, int out_size, void* d_ws, size_t ws_size,
                              hipStream_t stream) {
  const float* edge_val   = (const float*)d_in[0];
  const float* embed_user = (const float*)d_in[1];
  const float* embed_item = (const float*)d_in[2];
  const float* W1w        = (const float*)d_in[3];
  const float* W1b        = (const float*)d_in[4];
  const float* W2w        = (const float*)d_in[5];
  const float* W2b        = (const float*)d_in[6];
  const int*   erow       = (const int*)d_in[7];
  const int*   ecol       = (const int*)d_in[8];
  const int*   bu         = (const int*)d_in[9];
  const int*   bp         = (const int*)d_in[10];
  const int*   bn         = (const int*)d_in[11];
  float* out = (float*)d_out;

  const size_t NE = (size_t)N_NODES * EMBED;
  float* Ea = (float*)d_ws;
  float* Eb = Ea + NE;
  float* LE = Ea + 2 * NE;   // needs 3*NE*4 = ~115.2 MB of workspace

  const int n4 = N_NODES * 16;           // float4 count of E / LE
  k_init_E<<<(n4 + 255) / 256, 256, 0, stream>>>(
      (const float4*)embed_user, (const float4*)embed_item, (float4*)Ea);

  const int gthreads = 3 * BATCH * 16;
  k_gather<<<(gthreads + 255) / 256, 256, 0, stream>>>(Ea, 0, bu, bp, bn, out);

  float* cur = Ea;
  float* nxt = Eb;
  for (int l = 0; l < N_LAYERS; ++l) {
    k_zero<<<(n4 + 255) / 256, 256, 0, stream>>>((float4*)LE, n4);

    const int scatter_blocks = (NNZ * 32) / 256;   // 300000 blocks, exact
    k_scatter<<<scatter_blocks, 256, 0, stream>>>(edge_val, erow, ecol, cur, LE);

    k_layer<<<N_NODES / 16, 32, 0, stream>>>(
        LE, cur, W1w + (size_t)l * 64 * 64, W1b + (size_t)l * 64,
        W2w + (size_t)l * 64 * 64, W2b + (size_t)l * 64, nxt);

    k_gather<<<(gthreads + 255) / 256, 256, 0, stream>>>(nxt, l + 1, bu, bp, bn, out);

    float* tmp = cur; cur = nxt; nxt = tmp;
  }
}